// HighAccuracyGNN_25520695673306
// MI455X (gfx1250) — compile-verified
//
#include <hip/hip_runtime.h>
#include <hip/hip_bf16.h>

typedef float v2f __attribute__((ext_vector_type(2)));
typedef float v8f __attribute__((ext_vector_type(8)));

#define BM 64
#define BN 128
#define KC 16

// ---------------------------------------------------------------------------
// Dual-input fp32 WMMA GEMM:  D = (A1 * rowscale?) @ W1 [+ A2 @ W2] + bias
// A: M x K row-major, W: K x N row-major, D: M x N row-major.
// Block = 256 threads = 8 waves (4 M-subtiles x 2 N-subtiles); wave tile 16x64
// (4 accumulators). LDS panels are fragment-packed float2 along K so each
// lane's WMMA operand is a single 64-bit ds_load. Double-buffered: chunk i+1
// is fetched from global into registers while WMMAs consume buffer i.
// ---------------------------------------------------------------------------

// Per-thread register fetch of one K-chunk of staging data.
__device__ __forceinline__ void stage_fetch(
    const float* __restrict__ Ap, bool aOK, float sc,
    const float* __restrict__ Wp0, const float* __restrict__ Wp1,
    int k0, int N, bool nvec, int gcol0, int gcol1,
    float4& av, float4& w0, float4& w1)
{
    av = make_float4(0.f, 0.f, 0.f, 0.f);
    if (aOK) {
        av = *(const float4*)(Ap + k0);
        av.x *= sc; av.y *= sc; av.z *= sc; av.w *= sc;
    }
    w0 = make_float4(0.f, 0.f, 0.f, 0.f);
    w1 = make_float4(0.f, 0.f, 0.f, 0.f);
    const float* p0 = Wp0 + (size_t)k0 * N;
    const float* p1 = Wp1 + (size_t)k0 * N;
    if (nvec) {
        if (gcol0 + 3 < N) w0 = *(const float4*)p0;
        if (gcol1 + 3 < N) w1 = *(const float4*)p1;
    } else {
        if (gcol0 + 0 < N) w0.x = p0[0];
        if (gcol0 + 1 < N) w0.y = p0[1];
        if (gcol0 + 2 < N) w0.z = p0[2];
        if (gcol0 + 3 < N) w0.w = p0[3];
        if (gcol1 + 0 < N) w1.x = p1[0];
        if (gcol1 + 1 < N) w1.y = p1[1];
        if (gcol1 + 2 < N) w1.z = p1[2];
        if (gcol1 + 3 < N) w1.w = p1[3];
    }
}

__global__ __launch_bounds__(256)
void wmma_gemm_dual(const float* __restrict__ A1, const float* __restrict__ rowscale,
                    const float* __restrict__ W1,
                    const float* __restrict__ A2, const float* __restrict__ W2,
                    const float* __restrict__ bias,
                    float* __restrict__ D, int M, int K, int N)
{
    // As2[buf][r][p] = {A[r][2p], A[r][2p+1]}  (K-pair packed)
    // Bs2[buf][p][c] = {B[2p][c], B[2p+1][c]}
    __shared__ v2f As2[2][BM][KC / 2 + 1];   // 2 x 64 x 9   x 8B = 9.0 KB
    __shared__ v2f Bs2[2][KC / 2][BN + 2];   // 2 x 8  x 130 x 8B = 16.3 KB

    const int tid  = threadIdx.x;
    const int wave = tid >> 5;          // 0..7
    const int lane = tid & 31;
    const int wm   = wave >> 1;         // 0..3  (M sub-tile)
    const int wn   = wave & 1;          // 0..1  (N sub-tile, 64 cols each)
    const int g    = lane >> 4;         // lane half (ISA frag layout)
    const int l16  = lane & 15;

    const int blockM = blockIdx.x * BM;
    const int blockN = blockIdx.y * BN;

    // ---- invariant staging assignments (hoisted out of the K loop) ----
    const int aR    = tid >> 2;              // 0..63
    const int aC    = (tid & 3) * 4;         // 0,4,8,12
    const int aRowG = blockM + aR;
    const bool aOK  = (aRowG < M);
    const int bR0   = (tid * 4) >> 7;        // 0..7
    const int bC0   = (tid * 4) & 127;
    const int bR1   = (tid * 4 + 1024) >> 7; // 8..15
    const int gcol0 = blockN + bC0;
    const int gcol1 = gcol0;
    const bool nvec = ((N & 3) == 0);        // float4-safe W rows

    const int arow  = wm * 16 + l16;
    const int cbase = wn * 64 + l16;

    v8f acc0 = {}, acc1 = {}, acc2 = {}, acc3 = {};

    for (int pair = 0; pair < 2; ++pair) {
        const float* A = pair ? A2 : A1;
        const float* W = pair ? W2 : W1;
        if (A == nullptr) continue;

        const float* Ap  = A + (size_t)aRowG * K + aC;        // valid iff aOK
        const float* Wp0 = W + (size_t)bR0 * N + gcol0;
        const float* Wp1 = W + (size_t)bR1 * N + gcol1;
        const float sc = (pair == 0 && rowscale != nullptr && aOK)
                             ? rowscale[aRowG] : 1.0f;

        // ---- prologue: fetch + stage chunk 0 into buffer 0 ----
        float4 av, w0, w1;
        stage_fetch(Ap, aOK, sc, Wp0, Wp1, 0, N, nvec, gcol0, gcol1, av, w0, w1);
        {
            As2[0][aR][aC / 2 + 0] = v2f{av.x, av.y};
            As2[0][aR][aC / 2 + 1] = v2f{av.z, av.w};
            float* b0 = (float*)&Bs2[0][bR0 >> 1][bC0];
            b0[(bR0 & 1) + 0] = w0.x; b0[(bR0 & 1) + 2] = w0.y;
            b0[(bR0 & 1) + 4] = w0.z; b0[(bR0 & 1) + 6] = w0.w;
            float* b1 = (float*)&Bs2[0][bR1 >> 1][bC0];
            b1[(bR1 & 1) + 0] = w1.x; b1[(bR1 & 1) + 2] = w1.y;
            b1[(bR1 & 1) + 4] = w1.z; b1[(bR1 & 1) + 6] = w1.w;
        }

        int buf = 0;
        for (int k0 = 0; k0 < K; k0 += KC) {
            __syncthreads();                 // LDS buffer buf fully staged
            const bool more = (k0 + KC < K);
            if (more)                        // overlap: fetch next chunk now
                stage_fetch(Ap, aOK, sc, Wp0, Wp1, k0 + KC, N, nvec,
                            gcol0, gcol1, av, w0, w1);

            // ---- compute from LDS buffer buf ----
            #pragma unroll
            for (int kk = 0; kk < KC; kk += 4) {
                const int p = (kk >> 1) + g;
                v2f a  = As2[buf][arow][p];
                v2f b0 = Bs2[buf][p][cbase + 0];
                v2f b1 = Bs2[buf][p][cbase + 16];
                v2f b2 = Bs2[buf][p][cbase + 32];
                v2f b3 = Bs2[buf][p][cbase + 48];
                acc0 = __builtin_amdgcn_wmma_f32_16x16x4_f32(
                           false, a, false, b0, (short)0, acc0, false, false);
                acc1 = __builtin_amdgcn_wmma_f32_16x16x4_f32(
                           false, a, false, b1, (short)0, acc1, false, false);
                acc2 = __builtin_amdgcn_wmma_f32_16x16x4_f32(
                           false, a, false, b2, (short)0, acc2, false, false);
                acc3 = __builtin_amdgcn_wmma_f32_16x16x4_f32(
                           false, a, false, b3, (short)0, acc3, false, false);
            }

            if (more) {                      // stage chunk k0+KC into other buf
                const int nb = buf ^ 1;
                As2[nb][aR][aC / 2 + 0] = v2f{av.x, av.y};
                As2[nb][aR][aC / 2 + 1] = v2f{av.z, av.w};
                float* b0 = (float*)&Bs2[nb][bR0 >> 1][bC0];
                b0[(bR0 & 1) + 0] = w0.x; b0[(bR0 & 1) + 2] = w0.y;
                b0[(bR0 & 1) + 4] = w0.z; b0[(bR0 & 1) + 6] = w0.w;
                float* b1 = (float*)&Bs2[nb][bR1 >> 1][bC0];
                b1[(bR1 & 1) + 0] = w1.x; b1[(bR1 & 1) + 2] = w1.y;
                b1[(bR1 & 1) + 4] = w1.z; b1[(bR1 & 1) + 6] = w1.w;
                buf = nb;
            }
        }
        __syncthreads();   // pair boundary: all reads done before re-staging buf 0
    }

    // D layout per 16x16 tile: acc[v] holds (M = v + 8g, N = l16)
    #pragma unroll
    for (int v = 0; v < 8; ++v) {
        int row = blockM + wm * 16 + v + 8 * g;
        if (row >= M) continue;
        float* drow = D + (size_t)row * N;
        #pragma unroll
        for (int t = 0; t < 4; ++t) {
            int col = blockN + wn * 64 + t * 16 + l16;
            if (col < N) {
                float accv = (t == 0) ? acc0[v] : (t == 1) ? acc1[v]
                           : (t == 2) ? acc2[v] : acc3[v];
                drow[col] = accv + (bias ? bias[col] : 0.0f);
            }
        }
    }
}

// ---------------------------------------------------------------------------
// Edge scatter-sum: msum[dst] += H[src] (512 feats), cnt[dst] += 1.
// 2 edges per 256-thread block; 128 threads x float4 per edge.
// ---------------------------------------------------------------------------
__global__ __launch_bounds__(256)
void scatter_sum(const float* __restrict__ H, const int* __restrict__ src,
                 const int* __restrict__ dst, float* __restrict__ msum,
                 float* __restrict__ cnt, int nEdges, int N)
{
    int e = blockIdx.x * 2 + (threadIdx.x >> 7);
    if (e >= nEdges) return;
    int t = threadIdx.x & 127;            // 0..127, covers N=512 as float4
    int s = src[e];
    int d = dst[e];
    float4 v = ((const float4*)(H + (size_t)s * N))[t];
    float* out = msum + (size_t)d * N + t * 4;
    atomicAdd(out + 0, v.x);
    atomicAdd(out + 1, v.y);
    atomicAdd(out + 2, v.z);
    atomicAdd(out + 3, v.w);
    if (t == 0) atomicAdd(cnt + d, 1.0f);
}

__global__ __launch_bounds__(256)
void zero_f(float* __restrict__ p, size_t n)
{
    size_t i = (size_t)blockIdx.x * blockDim.x + threadIdx.x;
    if (i < n) p[i] = 0.0f;
}

__global__ __launch_bounds__(256)
void inv_cnt(const float* __restrict__ cnt, float* __restrict__ rs, int M)
{
    int i = blockIdx.x * blockDim.x + threadIdx.x;
    if (i < M) rs[i] = 1.0f / fmaxf(cnt[i], 1.0f);
}

// Column sums / sums-of-squares for GraphNorm (N fixed to 512).
__global__ __launch_bounds__(256)
void col_stats(const float* __restrict__ Z, float* __restrict__ csum,
               float* __restrict__ csq, int M)
{
    const int N = 512;
    int rbase = blockIdx.x * 64;
    int c = threadIdx.x;                  // 0..255 -> cols c and c+256
    float s0 = 0.f, s1 = 0.f, q0 = 0.f, q1 = 0.f;
    for (int r = 0; r < 64; ++r) {
        int row = rbase + r;
        if (row >= M) break;
        const float* zr = Z + (size_t)row * N;
        float a = zr[c];
        float b = zr[c + 256];
        s0 += a; q0 += a * a;
        s1 += b; q1 += b * b;
    }
    atomicAdd(&csum[c], s0);       atomicAdd(&csq[c], q0);
    atomicAdd(&csum[c + 256], s1); atomicAdd(&csq[c + 256], q1);
}

// Collapse GraphNorm to y = scale*x + shift per channel.
// mu = E[x]; var = E[x^2] - 2*a*mu^2 + a^2*mu^2; scale = g*rsqrt(var+eps);
// shift = b - scale*a*mu.
__global__ void norm_prep(const float* __restrict__ csum, const float* __restrict__ csq,
                          const float* __restrict__ gw, const float* __restrict__ gb,
                          const float* __restrict__ alpha,
                          float* __restrict__ scale, float* __restrict__ shift, int M)
{
    int j = threadIdx.x;                  // 512 threads
    float invM = 1.0f / (float)M;
    float mu  = csum[j] * invM;
    float ex2 = csq[j] * invM;
    float al  = alpha[j];
    float var = ex2 - 2.0f * al * mu * mu + al * al * mu * mu;
    float sc  = gw[j] * rsqrtf(var + 1e-5f);
    scale[j] = sc;
    shift[j] = gb[j] - sc * al * mu;
}

// Fused GraphNorm apply + LeakyReLU(0.1), float4 vectorized. N fixed 512.
__global__ __launch_bounds__(256)
void norm_apply(const float* __restrict__ Z, const float* __restrict__ scale,
                const float* __restrict__ shift, float* __restrict__ H, size_t total4)
{
    const int N = 512;
    size_t i = (size_t)blockIdx.x * blockDim.x + threadIdx.x;
    if (i >= total4) return;
    int c = (int)((i * 4) & (N - 1));
    float4 z = ((const float4*)Z)[i];
    float4 y;
    y.x = scale[c + 0] * z.x + shift[c + 0];
    y.y = scale[c + 1] * z.y + shift[c + 1];
    y.z = scale[c + 2] * z.z + shift[c + 2];
    y.w = scale[c + 3] * z.w + shift[c + 3];
    y.x = y.x > 0.f ? y.x : 0.1f * y.x;
    y.y = y.y > 0.f ? y.y : 0.1f * y.y;
    y.z = y.z > 0.f ? y.z : 0.1f * y.z;
    y.w = y.w > 0.f ? y.w : 0.1f * y.w;
    ((float4*)H)[i] = y;
}

// ---------------------------------------------------------------------------
extern "C" void kernel_launch(void* const* d_in, const int* in_sizes, int n_in,
                              void* d_out, int out_size, void* d_ws, size_t ws_size,
                              hipStream_t stream)
{
    const int IN_CH = 256, HID = 512, OUT_CH = 121;
    const int M = in_sizes[0] / IN_CH;        // 50000 nodes
    const int E = in_sizes[1] / 2;            // 800000 edges

    const float* x     = (const float*)d_in[0];
    const int*   eidx  = (const int*)d_in[1]; // jax int64 request -> int32 materialized
    const int*   src   = eidx;
    const int*   dst   = eidx + E;
    const float* W_in  = (const float*)d_in[2];
    const float* b_in  = (const float*)d_in[3];
    const float* Wl[3] = { (const float*)d_in[4],  (const float*)d_in[10], (const float*)d_in[16] };
    const float* bl[3] = { (const float*)d_in[5],  (const float*)d_in[11], (const float*)d_in[17] };
    const float* Wr[3] = { (const float*)d_in[6],  (const float*)d_in[12], (const float*)d_in[18] };
    const float* gg[3] = { (const float*)d_in[7],  (const float*)d_in[13], (const float*)d_in[19] };
    const float* be[3] = { (const float*)d_in[8],  (const float*)d_in[14], (const float*)d_in[20] };
    const float* al[3] = { (const float*)d_in[9],  (const float*)d_in[15], (const float*)d_in[21] };
    const float* W_out = (const float*)d_in[22];
    const float* b_out = (const float*)d_in[23];
    float* out = (float*)d_out;

    // workspace layout (floats)
    float* ws = (float*)d_ws;
    const size_t MH = (size_t)M * HID;
    float* Z     = ws;
    float* H     = Z + MH;
    float* MS    = H + MH;
    float* cnt   = MS + MH;
    float* rs    = cnt + M;
    float* csum  = rs + M;
    float* csq   = csum + HID;
    float* scale = csq + HID;
    float* shift = scale + HID;

    const dim3 blk(256);
    const int gM = (M + BM - 1) / BM;                     // 782
    const dim3 gemmGridH(gM, HID / BN);                   // N=512 -> 4
    const dim3 gemmGridO(gM, (OUT_CH + BN - 1) / BN);     // N=121 -> 1
    const int statGrid = (M + 63) / 64;
    const size_t total4 = MH / 4;
    const int applyGrid = (int)((total4 + 255) / 256);

    // fc_in: H = x @ W_in + b_in
    wmma_gemm_dual<<<gemmGridH, blk, 0, stream>>>(
        x, nullptr, W_in, nullptr, nullptr, b_in, H, M, IN_CH, HID);

    for (int L = 0; L < 3; ++L) {
        // zero accumulators: MS + cnt (contiguous), then csum+csq (contiguous)
        zero_f<<<(int)((MH + M + 255) / 256), blk, 0, stream>>>(MS, MH + (size_t)M);
        zero_f<<<(2 * HID + 255) / 256, blk, 0, stream>>>(csum, (size_t)(2 * HID));

        // mean aggregation (sum + counts)
        scatter_sum<<<(E + 1) / 2, blk, 0, stream>>>(H, src, dst, MS, cnt, E, HID);
        inv_cnt<<<(M + 255) / 256, blk, 0, stream>>>(cnt, rs, M);

        // Z = (MS * rs) @ Wl + H @ Wr + bl   (mean fused as row scale)
        wmma_gemm_dual<<<gemmGridH, blk, 0, stream>>>(
            MS, rs, Wl[L], H, Wr[L], bl[L], Z, M, HID, HID);

        // GraphNorm + LeakyReLU -> H
        col_stats<<<statGrid, blk, 0, stream>>>(Z, csum, csq, M);
        norm_prep<<<1, HID, 0, stream>>>(csum, csq, gg[L], be[L], al[L], scale, shift, M);
        norm_apply<<<applyGrid, blk, 0, stream>>>(Z, scale, shift, H, total4);
    }

    // fc_out: out = H @ W_out + b_out
    wmma_gemm_dual<<<gemmGridO, blk, 0, stream>>>(
        H, nullptr, W_out, nullptr, nullptr, b_out, out, M, HID, OUT_CH);
}